// Local_39152921870765
// MI455X (gfx1250) — compile-verified
//
#include <hip/hip_runtime.h>
#include <hip/hip_bf16.h>

typedef __attribute__((ext_vector_type(2))) float v2f;
typedef __attribute__((ext_vector_type(8))) float v8f;

#define IN_LEN  65536
#define OUT_CH  512
#define KERNEL  32
#define STRIDE  16
#define PAD     8
#define FOLD    4096
#define BATCH   64

// One workgroup (8 waves) per fold f.
//   A_f (64x32)  = x windows, staged in LDS (padded rows: 33 floats, conflict-free)
//   B_f (32x512) = weight[f], streamed from global (zero reuse across folds)
//   C_f (64x512) = out[:, f, :]
// Wave w computes N columns [w*64, w*64+64): 4 N-tiles x 4 M-tiles x 8 k-steps
// of V_WMMA_F32_16X16X4_F32.
__global__ __launch_bounds__(256) void Local_39152921870765_fold_gemm(
    const float* __restrict__ x,
    const float* __restrict__ w,
    float* __restrict__ out)
{
    const int f = blockIdx.x;

    __shared__ float As[BATCH][KERNEL + 1];   // +1 pad: row stride 33 floats

    const int tid = threadIdx.x;              // 0..255
    const int start = f * STRIDE - PAD;       // window start in x (may be <0)

    // Cooperative load of the 64x32 window tile (2048 elems, 8 per thread).
    #pragma unroll
    for (int i = tid; i < BATCH * KERNEL; i += 256) {
        const int b = i >> 5;                 // batch row
        const int k = i & 31;                 // position in kernel window
        const int src = start + k;
        float v = 0.0f;
        if (src >= 0 && src < IN_LEN) v = x[(size_t)b * IN_LEN + src];
        As[b][k] = v;
    }
    __syncthreads();

    const int wave = tid >> 5;                // 0..7
    const int lane = tid & 31;
    const int half = lane >> 4;               // 0: lanes 0-15, 1: lanes 16-31
    const int l16  = lane & 15;

    const float* __restrict__ wf = w + (size_t)f * (KERNEL * OUT_CH);
    float* __restrict__ of = out + (size_t)f * OUT_CH;

    const int nWave = wave * 64;              // this wave's N base

    #pragma unroll
    for (int nt = 0; nt < 4; ++nt) {
        const int col = nWave + nt * 16 + l16;    // N = lane&15 within tile

        // B fragments for all 8 k-steps of this column tile.
        // Layout (ISA 7.12.2, 32-bit 4x16): VGPR0 holds K = 2*half,
        // VGPR1 holds K = 2*half+1, N = lane&15.
        v2f bfrag[8];
        #pragma unroll
        for (int kk = 0; kk < 8; ++kk) {
            const int kr = kk * 4 + half * 2;
            bfrag[kk].x = wf[(size_t)kr       * OUT_CH + col];
            bfrag[kk].y = wf[(size_t)(kr + 1) * OUT_CH + col];
        }

        #pragma unroll
        for (int mt = 0; mt < 4; ++mt) {
            v8f acc = {};
            #pragma unroll
            for (int kk = 0; kk < 8; ++kk) {
                // A fragment (ISA 7.12.2, 32-bit 16x4): M = lane&15,
                // VGPR0 = K(2*half), VGPR1 = K(2*half+1).
                const int m  = mt * 16 + l16;
                const int kr = kk * 4 + half * 2;
                v2f afrag;
                afrag.x = As[m][kr];
                afrag.y = As[m][kr + 1];
                acc = __builtin_amdgcn_wmma_f32_16x16x4_f32(
                    /*neg_a=*/false, afrag,
                    /*neg_b=*/false, bfrag[kk],
                    /*c_mod=*/(short)0, acc,
                    /*reuse_a=*/false, /*reuse_b=*/false);
            }

            // C/D layout (ISA 7.12.2, 32-bit 16x16): VGPR i ->
            // lanes 0-15: (M = i,   N = lane), lanes 16-31: (M = 8+i, N = lane-16)
            const int n    = nWave + nt * 16 + l16;
            const int mrow = mt * 16 + half * 8;
            #pragma unroll
            for (int i = 0; i < 8; ++i) {
                const int b = mrow + i;
                of[(size_t)b * (FOLD * OUT_CH) + n] = acc[i];
            }
        }
    }
}

extern "C" void kernel_launch(void* const* d_in, const int* in_sizes, int n_in,
                              void* d_out, int out_size, void* d_ws, size_t ws_size,
                              hipStream_t stream) {
    const float* x = (const float*)d_in[0];   // (64, 65536) f32
    const float* w = (const float*)d_in[1];   // (4096, 32, 512) f32
    float* out = (float*)d_out;               // (64, 4096, 512) f32

    dim3 grid(FOLD);
    dim3 block(256);
    Local_39152921870765_fold_gemm<<<grid, block, 0, stream>>>(x, w, out);
}